// ArrayDecoderWithHistory_56762287784319
// MI455X (gfx1250) — compile-verified
//
#include <hip/hip_runtime.h>
#include <math.h>

// ---------------------------------------------------------------------------
// Fully fused spectral decoder for MI455X (gfx1250, wave32, WMMA).
// One 256-thread block (8 waves) handles one (batch, 32-token) tile and runs
// the entire network (7x [conv -> LN -> LN -> FFN -> proj] -> concat -> mix
// MLP) with all intermediates in dynamic LDS (~124 KB; 2 blocks fit in the
// 320 KB WGP LDS). Each wave owns one 16-wide N-tile and computes BOTH
// 16-row M-tiles, reusing the B-operand registers across two
// V_WMMA_F32_16X16X4_F32 issues -> halves L2 weight traffic per token.
// ---------------------------------------------------------------------------

#define NBANDS   7
#define SDIM     16     // input / output feature dim
#define DDIM     128    // band model dim
#define BATCH    32
#define TLEN     2048
#define KMAXC    31     // conv kernel (zero-padded)
#define HALO     15     // KMAXC/2
#define TILE_T   32     // tokens per block (2 WMMA M-tiles)
#define D2       256    // FFN hidden
#define PROJ     64     // 4*S per-band projection
#define CATD     448    // NBANDS * PROJ
#define MIXH     128    // 8*S mix hidden

// LDS carve-up (floats)
#define XBUF_N   ((TILE_T + 2 * HALO) * SDIM)   // 992
#define HBUF_OFF (XBUF_N)                        // 992
#define ZBUF_OFF (HBUF_OFF + TILE_T * DDIM)      // 5088
#define UBUF_OFF (ZBUF_OFF + TILE_T * DDIM)      // 9184
#define CAT_OFF  (UBUF_OFF + TILE_T * D2)        // 17376
#define SMEM_FLTS (CAT_OFF + TILE_T * CATD)      // 31712 -> 126848 B

typedef float v2f __attribute__((ext_vector_type(2)));
typedef float v8f __attribute__((ext_vector_type(8)));

// D = A(16x4) * B(4x16) + C   (f32, wave32, per CDNA5 ISA)
__device__ __forceinline__ v8f wmma4(v2f a, v2f b, v8f c) {
  // 8 args: (neg_a, A, neg_b, B, c_mod, C, reuse_a, reuse_b)
  return __builtin_amdgcn_wmma_f32_16x16x4_f32(false, a, false, b,
                                               (short)0, c, false, false);
}

__device__ __forceinline__ float gelu_exact(float x) {
  return 0.5f * x * (1.0f + erff(x * 0.70710678118654752440f));
}

// Two-M-tile GEMM: A [32 x ktot] row-major in LDS (row stride lda; M-tile 1 at
// rows 16..31), B [ktot x *] row-major global (pre-offset to this tile's
// column base, row stride ldb). B operand loaded ONCE per K-step and reused
// by both WMMAs. A-operand layout (32-bit A 16x4): lanes 0-15 hold K={k,k+1}
// of row M=lane; lanes 16-31 hold K={k+2,k+3} of row M=lane-16. B mirrored.
__device__ __forceinline__ void gemm_tile2(const float* __restrict__ a, int lda,
                                           const float* __restrict__ b, int ldb,
                                           int ktot, int lane,
                                           v8f& acc0, v8f& acc1) {
  const int arow = lane & 15;
  const int koff = (lane >> 4) << 1;   // 0 for lanes 0-15, 2 for lanes 16-31
  const int bcol = lane & 15;
#pragma unroll 4
  for (int kk = 0; kk < ktot; kk += 4) {
    const float* bp = b + (kk + koff) * ldb + bcol;
    v2f bv; bv[0] = bp[0]; bv[1] = bp[ldb];            // shared B operand
    const float* ap0 = a + arow * lda + kk + koff;
    const float* ap1 = ap0 + 16 * lda;                  // rows 16..31
    v2f a0; a0[0] = ap0[0]; a0[1] = ap0[1];             // ds_load_b64
    v2f a1; a1[0] = ap1[0]; a1[1] = ap1[1];
    acc0 = wmma4(a0, bv, acc0);
    acc1 = wmma4(a1, bv, acc1);
  }
}

// Store a 16x16 C/D tile (vgpr r -> row r, lanes>=16 -> rows 8..15) into a
// row-major LDS buffer (dst pre-offset to tile row/column base).
__device__ __forceinline__ void store_tile(float* __restrict__ dst, int ldd,
                                           v8f acc,
                                           const float* __restrict__ bias,
                                           int lane, bool act, bool residual) {
  const int c = lane & 15;
  const int rbase = (lane >> 4) << 3;  // 0 or 8
  const float bv = bias[c];
#pragma unroll
  for (int r = 0; r < 8; ++r) {
    float v = acc[r] + bv;
    if (act) v = gelu_exact(v);
    if (residual) dst[(rbase + r) * ldd + c] += v;
    else          dst[(rbase + r) * ldd + c]  = v;
  }
}

__device__ __forceinline__ void store_tile2(float* __restrict__ dst, int ldd,
                                            v8f acc0, v8f acc1,
                                            const float* __restrict__ bias,
                                            int lane, bool act, bool residual) {
  store_tile(dst,            ldd, acc0, bias, lane, act, residual);
  store_tile(dst + 16 * ldd, ldd, acc1, bias, lane, act, residual);
}

// Cooperative LayerNorm over D=128 for 16 rows: 16 lanes per row (threads
// row*16 .. row*16+15 sit in the same wave), shfl-xor reduction width 16.
__device__ __forceinline__ void layernorm_16x128(const float* __restrict__ src,
                                                 float* __restrict__ dst,
                                                 const float* __restrict__ g,
                                                 const float* __restrict__ b,
                                                 int tid) {
  const int row = tid >> 4;
  const int l16 = tid & 15;
  float vals[8];
  float s = 0.f;
#pragma unroll
  for (int j = 0; j < 8; ++j) {
    vals[j] = src[row * DDIM + l16 + 16 * j];
    s += vals[j];
  }
#pragma unroll
  for (int m = 8; m >= 1; m >>= 1) s += __shfl_xor(s, m, 16);
  const float mean = s * (1.0f / DDIM);
  float vs = 0.f;
#pragma unroll
  for (int j = 0; j < 8; ++j) { float d = vals[j] - mean; vs += d * d; }
#pragma unroll
  for (int m = 8; m >= 1; m >>= 1) vs += __shfl_xor(vs, m, 16);
  const float rstd = rsqrtf(vs * (1.0f / DDIM) + 1e-5f);
#pragma unroll
  for (int j = 0; j < 8; ++j) {
    const int c = l16 + 16 * j;
    dst[row * DDIM + c] = (vals[j] - mean) * rstd * g[c] + b[c];
  }
}

__global__ __launch_bounds__(256)
void fused_spectral_decoder(const float* __restrict__ x,
                            const float* __restrict__ conv_w,
                            const float* __restrict__ conv_b,
                            const float* __restrict__ dec_g,
                            const float* __restrict__ dec_b,
                            const float* __restrict__ n2_g,
                            const float* __restrict__ n2_b,
                            const float* __restrict__ ffn_w1,
                            const float* __restrict__ ffn_b1,
                            const float* __restrict__ ffn_w2,
                            const float* __restrict__ ffn_b2,
                            const float* __restrict__ proj_w,
                            const float* __restrict__ proj_b,
                            const float* __restrict__ mix_w1,
                            const float* __restrict__ mix_b1,
                            const float* __restrict__ mix_w2,
                            const float* __restrict__ mix_b2,
                            float* __restrict__ out) {
  extern __shared__ float smem[];
  float* xbuf   = smem;              // [62 x 16] im2col window
  float* hbuf   = smem + HBUF_OFF;   // [32 x 128] h / band_out
  float* zbuf   = smem + ZBUF_OFF;   // [32 x 128] LN2 output
  float* ubuf   = smem + UBUF_OFF;   // [32 x 256] FFN hidden; reused as mix hidden
  float* catbuf = smem + CAT_OFF;    // [32 x 448] concatenated band projections

  const int tid  = threadIdx.x;
  const int lane = tid & 31;
  const int wave = tid >> 5;
  const int nTileT = TLEN / TILE_T;                  // 64
  const int bb = blockIdx.x / nTileT;
  const int t0 = (blockIdx.x % nTileT) * TILE_T;

  // --- load x patch (with halo, zero-padded) -------------------------------
  for (int i = tid; i < XBUF_N; i += 256) {
    const int tok = t0 - HALO + (i >> 4);
    const int s   = i & 15;
    xbuf[i] = (tok >= 0 && tok < TLEN) ? x[(bb * TLEN + tok) * SDIM + s] : 0.0f;
  }
  __syncthreads();

  for (int n = 0; n < NBANDS; ++n) {
    const float* cw = conv_w + (size_t)n * KMAXC * SDIM * DDIM; // [496 x 128]
    if (n + 1 < NBANDS)
      __builtin_prefetch(conv_w + (size_t)(n + 1) * KMAXC * SDIM * DDIM, 0, 1);

    // --- conv as GEMM: A rows t = xbuf + 16t (overlapping im2col), K=496 ---
    {
      const int n0 = wave * 16;                       // 8 N-tiles of 128
      v8f acc0 = {}, acc1 = {};
      gemm_tile2(xbuf, SDIM, cw + n0, DDIM, KMAXC * SDIM, lane, acc0, acc1);
      store_tile2(hbuf + n0, DDIM, acc0, acc1, conv_b + n * DDIM + n0, lane,
                  /*act=*/false, /*residual=*/false);
    }
    __syncthreads();

    // --- LN1 (dec) in-place, LN2 (n2) -> zbuf, both 16-row halves ----------
    layernorm_16x128(hbuf,             hbuf,             dec_g + n * DDIM, dec_b + n * DDIM, tid);
    layernorm_16x128(hbuf + 16 * DDIM, hbuf + 16 * DDIM, dec_g + n * DDIM, dec_b + n * DDIM, tid);
    __syncthreads();
    layernorm_16x128(hbuf,             zbuf,             n2_g + n * DDIM, n2_b + n * DDIM, tid);
    layernorm_16x128(hbuf + 16 * DDIM, zbuf + 16 * DDIM, n2_g + n * DDIM, n2_b + n * DDIM, tid);
    __syncthreads();

    // --- FFN1: z[32x128] @ w1[128x256] + b1, GELU -> ubuf ------------------
    {
      const float* w1 = ffn_w1 + (size_t)n * DDIM * D2;
#pragma unroll
      for (int half = 0; half < 2; ++half) {
        const int n0 = (wave + half * 8) * 16;        // 16 N-tiles of 256
        v8f acc0 = {}, acc1 = {};
        gemm_tile2(zbuf, DDIM, w1 + n0, D2, DDIM, lane, acc0, acc1);
        store_tile2(ubuf + n0, D2, acc0, acc1, ffn_b1 + n * D2 + n0, lane,
                    /*act=*/true, /*residual=*/false);
      }
    }
    __syncthreads();

    // --- FFN2: u[32x256] @ w2[256x128] + b2, residual into hbuf ------------
    {
      const float* w2 = ffn_w2 + (size_t)n * D2 * DDIM;
      const int n0 = wave * 16;                       // 8 N-tiles
      v8f acc0 = {}, acc1 = {};
      gemm_tile2(ubuf, D2, w2 + n0, DDIM, D2, lane, acc0, acc1);
      store_tile2(hbuf + n0, DDIM, acc0, acc1, ffn_b2 + n * DDIM + n0, lane,
                  /*act=*/false, /*residual=*/true);
    }
    __syncthreads();

    // --- proj: band_out[32x128] @ proj_w[128x64] + b -> catbuf cols n*64 ---
    if (wave < 4) {
      const float* pw = proj_w + (size_t)n * DDIM * PROJ;
      const int n0 = wave * 16;                       // 4 N-tiles of 64
      v8f acc0 = {}, acc1 = {};
      gemm_tile2(hbuf, DDIM, pw + n0, PROJ, DDIM, lane, acc0, acc1);
      store_tile2(catbuf + n * PROJ + n0, CATD, acc0, acc1,
                  proj_b + n * PROJ + n0, lane,
                  /*act=*/false, /*residual=*/false);
    }
    __syncthreads();
  }

  // --- mix1: cat[32x448] @ mix_w1[448x128] + b1, GELU -> ubuf (reused) -----
  {
    const int n0 = wave * 16;                         // 8 N-tiles of 128
    v8f acc0 = {}, acc1 = {};
    gemm_tile2(catbuf, CATD, mix_w1 + n0, MIXH, CATD, lane, acc0, acc1);
    store_tile2(ubuf + n0, MIXH, acc0, acc1, mix_b1 + n0, lane,
                /*act=*/true, /*residual=*/false);
  }
  __syncthreads();

  // --- mix2: m[32x128] @ mix_w2[128x16] + b2 -> out ------------------------
  if (wave == 0) {
    v8f acc0 = {}, acc1 = {};
    gemm_tile2(ubuf, MIXH, mix_w2, SDIM, MIXH, lane, acc0, acc1);
    const int c = lane & 15;
    const int rbase = (lane >> 4) << 3;
    const float bv = mix_b2[c];
#pragma unroll
    for (int r = 0; r < 8; ++r) {
      out[((size_t)bb * TLEN + t0 + rbase + r) * SDIM + c]      = acc0[r] + bv;
      out[((size_t)bb * TLEN + t0 + 16 + rbase + r) * SDIM + c] = acc1[r] + bv;
    }
  }
}

extern "C" void kernel_launch(void* const* d_in, const int* in_sizes, int n_in,
                              void* d_out, int out_size, void* d_ws,
                              size_t ws_size, hipStream_t stream) {
  (void)in_sizes; (void)n_in; (void)d_ws; (void)ws_size; (void)out_size;
  const float* x       = (const float*)d_in[0];
  const float* conv_w  = (const float*)d_in[1];
  const float* conv_b  = (const float*)d_in[2];
  const float* dec_g   = (const float*)d_in[3];
  const float* dec_b   = (const float*)d_in[4];
  const float* n2_g    = (const float*)d_in[5];
  const float* n2_b    = (const float*)d_in[6];
  const float* ffn_w1  = (const float*)d_in[7];
  const float* ffn_b1  = (const float*)d_in[8];
  const float* ffn_w2  = (const float*)d_in[9];
  const float* ffn_b2  = (const float*)d_in[10];
  const float* proj_w  = (const float*)d_in[11];
  const float* proj_b  = (const float*)d_in[12];
  const float* mix_w1  = (const float*)d_in[13];
  const float* mix_b1  = (const float*)d_in[14];
  const float* mix_w2  = (const float*)d_in[15];
  const float* mix_b2  = (const float*)d_in[16];
  float* out = (float*)d_out;

  const size_t smemBytes = (size_t)SMEM_FLTS * sizeof(float);  // 126848 B
  // Allow >64KB dynamic LDS (320KB per WGP on gfx1250).
  (void)hipFuncSetAttribute((const void*)fused_spectral_decoder,
                            hipFuncAttributeMaxDynamicSharedMemorySize,
                            (int)smemBytes);

  const int grid = BATCH * (TLEN / TILE_T);          // 2048 blocks
  fused_spectral_decoder<<<dim3(grid), dim3(256), smemBytes, stream>>>(
      x, conv_w, conv_b, dec_g, dec_b, n2_g, n2_b,
      ffn_w1, ffn_b1, ffn_w2, ffn_b2,
      proj_w, proj_b, mix_w1, mix_b1, mix_w2, mix_b2, out);
}